// GCN_80650895884394
// MI455X (gfx1250) — compile-verified
//
#include <hip/hip_runtime.h>
#include <hip/hip_bf16.h>

typedef float v2f __attribute__((ext_vector_type(2)));
typedef float v8f __attribute__((ext_vector_type(8)));
typedef unsigned int u32x4 __attribute__((ext_vector_type(4)));
typedef int i32x4 __attribute__((ext_vector_type(4)));
typedef int i32x8 __attribute__((ext_vector_type(8)));

// ---------------------------------------------------------------- utilities
__global__ void zero_f32(float* __restrict__ p, long long n) {
    long long i = (long long)blockIdx.x * blockDim.x + threadIdx.x;
    long long stride = (long long)gridDim.x * blockDim.x;
    for (; i < n; i += stride) p[i] = 0.0f;
}

__global__ void deg_kernel(const long long* __restrict__ dst, float* __restrict__ deg, int E) {
    int e = blockIdx.x * blockDim.x + threadIdx.x;
    if (e < E) atomicAdd(&deg[(int)dst[e]], 1.0f);
}

__global__ void dinv_kernel(const float* __restrict__ deg, float* __restrict__ dinv, int N) {
    int i = blockIdx.x * blockDim.x + threadIdx.x;
    if (i < N) dinv[i] = __frsqrt_rn(fmaxf(deg[i], 1.0f));
}

__global__ void norm_kernel(const long long* __restrict__ src, const long long* __restrict__ dst,
                            const float* __restrict__ dinv, float* __restrict__ norm, int E) {
    int e = blockIdx.x * blockDim.x + threadIdx.x;
    if (e < E) norm[e] = dinv[(int)src[e]] * dinv[(int)dst[e]];
}

// ------------------------------------------------- SpMM scatter (F == 128)
// One edge per wave32; each lane moves a float4 (32 lanes * 4 = 128 floats).
__global__ void spmm_scatter128(const float* __restrict__ X,
                                const long long* __restrict__ src,
                                const long long* __restrict__ dst,
                                const float* __restrict__ norm,
                                float* __restrict__ agg, int E) {
    long long t = (long long)blockIdx.x * blockDim.x + threadIdx.x;
    long long e = t >> 5;
    int lane = (int)(t & 31);
    if (e >= E) return;
    int s = (int)src[e];
    int d = (int)dst[e];
    float nv = norm[e];
    const float4 x = ((const float4*)(X + (size_t)s * 128))[lane];
    float* out = agg + (size_t)d * 128 + lane * 4;
    atomicAdd(out + 0, x.x * nv);
    atomicAdd(out + 1, x.y * nv);
    atomicAdd(out + 2, x.z * nv);
    atomicAdd(out + 3, x.w * nv);
}

// ------------------------------------------------ TDM: stage W[K x Fout] into LDS
// D# bitfields per CDNA5 ISA. One wave issues the DMA
// (EXEC ignored), waits TENSORcnt==0; caller barriers before LDS is read.
__device__ __forceinline__ void tdm_stage_W(const float* W, float* sW, int Fout, int K) {
#if __has_builtin(__builtin_amdgcn_tensor_load_to_lds)
    unsigned long long ga = (unsigned long long)(uintptr_t)W;
    unsigned int lds = (unsigned int)(uintptr_t)sW;   // flat addr low 32b == LDS offset
    u32x4 g0;
    g0.x = 1u;                                         // count=1, user-mode, no gather
    g0.y = lds;                                        // lds_addr
    g0.z = (unsigned int)(ga & 0xFFFFFFFFu);           // global_addr[31:0]
    g0.w = (unsigned int)((ga >> 32) & 0x01FFFFFFu)    // global_addr[56:32]
           | (2u << 30);                               // type=2 ("image")
    i32x8 g1;
    g1[0] = (int)(2u << 16);                           // data_size=2 (4 bytes); mask=0
    g1[1] = (int)((unsigned)Fout << 16);               // tensor_dim0[15:0]
    g1[2] = (int)((unsigned)K << 16);                  // tensor_dim0[31:16]=0 | tensor_dim1[15:0]
    g1[3] = (int)((unsigned)Fout << 16);               // tensor_dim1 hi=0 | tile_dim0
    g1[4] = (int)(unsigned)K;                          // tile_dim1 | tile_dim2=0
    g1[5] = (int)(unsigned)Fout;                       // tensor_dim0_stride[31:0]
    g1[6] = 0;                                         // stride0 hi | dim1_stride lo
    g1[7] = 0;
    i32x4 gz = {0, 0, 0, 0};
#if __clang_major__ >= 23
    i32x8 gz8 = {0, 0, 0, 0, 0, 0, 0, 0};
    __builtin_amdgcn_tensor_load_to_lds(g0, g1, gz, gz, gz8, 0);
#else
    __builtin_amdgcn_tensor_load_to_lds(g0, g1, gz, gz, 0);
#endif
    __builtin_amdgcn_s_wait_tensorcnt(0);
#else
    for (int i = threadIdx.x & 31; i < K * Fout; i += 32) sW[i] = W[i];
#endif
}

// -------------------------------------------- fp32 WMMA GEMM: H = relu(A@W)
// A:[N,128], W:[128,Fout] staged in LDS via TDM, H:[N,Fout]. Each wave owns a
// 16-row strip and all TILES column tiles, reusing its A fragment TILES times.
// Operand layouts per CDNA5 ISA (A 16x4: lanes 0-15 K0/K1, lanes
// 16-31 K2/K3; B mirrored; C/D 8 VGPRs).
template <int TILES>
__global__ void gemm_relu_wmma_lds(const float* __restrict__ A, const float* __restrict__ W,
                                   float* __restrict__ H, int N) {
    constexpr int Fout = TILES * 16;
    constexpr int K = 128;
    __shared__ float sW[K * Fout];

    const int wave = threadIdx.x >> 5;
    const int lane = threadIdx.x & 31;

    if (wave == 0) tdm_stage_W(W, sW, Fout, K);
    __syncthreads();

    const int row0 = (blockIdx.x * (blockDim.x >> 5) + wave) << 4;
    if (row0 < N) {                                    // wave-uniform: EXEC all-1s
        const int m16 = lane & 15;                     // M (A) / col (B) in tile
        const int khi = (lane >> 4) << 1;              // 0 | 2
        const float* arow = A + (size_t)(row0 + m16) * K + khi;

        v8f acc[TILES];
#pragma unroll
        for (int t = 0; t < TILES; ++t) acc[t] = (v8f){};

        for (int k = 0; k < K; k += 4) {
            v2f a = *(const v2f*)(arow + k);           // A[M][k+khi .. +1], 8B aligned
            const float* b0 = &sW[(k + khi) * Fout + m16];
#pragma unroll
            for (int t = 0; t < TILES; ++t) {
                v2f b;
                b.x = b0[t * 16];                      // W[k+khi  ][col] (LDS)
                b.y = b0[t * 16 + Fout];               // W[k+khi+1][col] (LDS)
                acc[t] = __builtin_amdgcn_wmma_f32_16x16x4_f32(
                    false, a, false, b, (short)0, acc[t], false, false);
            }
        }

        const int mbase = row0 + ((lane >> 4) << 3);
#pragma unroll
        for (int t = 0; t < TILES; ++t) {
            const int col = t * 16 + m16;
#pragma unroll
            for (int r = 0; r < 8; ++r) {
                H[(size_t)(mbase + r) * Fout + col] = fmaxf(acc[t][r], 0.0f);
            }
        }
    }
}

// --------------------------------------------- softmax over rows of 64, in place
__global__ void softmax64(float* __restrict__ H, int N) {
    const int wave = threadIdx.x >> 5;
    const int lane = threadIdx.x & 31;
    const int row = blockIdx.x * (blockDim.x >> 5) + wave;
    if (row >= N) return;
    float* p = H + (size_t)row * 64;
    float a = p[lane];
    float b = p[lane + 32];
    float m = fmaxf(a, b);
    for (int off = 16; off > 0; off >>= 1) m = fmaxf(m, __shfl_xor(m, off, 32));
    float ea = __expf(a - m);
    float eb = __expf(b - m);
    float s = ea + eb;
    for (int off = 16; off > 0; off >>= 1) s += __shfl_xor(s, off, 32);
    float inv = 1.0f / s;
    p[lane] = ea * inv;
    p[lane + 32] = eb * inv;
}

// ---------------------------------------------------------------- launcher
extern "C" void kernel_launch(void* const* d_in, const int* in_sizes, int n_in,
                              void* d_out, int out_size, void* d_ws, size_t ws_size,
                              hipStream_t stream) {
    const float* X      = (const float*)d_in[0];
    const long long* ei = (const long long*)d_in[1];
    const float* W1     = (const float*)d_in[2];
    const float* W2     = (const float*)d_in[3];
    const float* W3     = (const float*)d_in[4];
    float* out = (float*)d_out;

    const int N = in_sizes[0] / 128;   // 50000
    const int E = in_sizes[1] / 2;     // 800000
    const long long* src = ei;
    const long long* dst = ei + E;

    // workspace layout (floats)
    float* ws   = (float*)d_ws;
    float* deg  = ws;                              // N
    float* dinv = deg + N;                         // N
    float* norm = dinv + N;                        // E
    float* agg  = norm + E;                        // N*128 (16B aligned)
    float* h    = agg + (size_t)N * 128;           // N*128

    const int B = 256;
    const int wavesPerBlock = B / 32;
    const int rowsPerBlock = wavesPerBlock * 16;   // 128
    const int gemmBlocks = (N + rowsPerBlock - 1) / rowsPerBlock;

    // degrees + normalization
    zero_f32<<<2048, B, 0, stream>>>(deg, N);
    deg_kernel<<<(E + B - 1) / B, B, 0, stream>>>(dst, deg, E);
    dinv_kernel<<<(N + B - 1) / B, B, 0, stream>>>(deg, dinv, N);
    norm_kernel<<<(E + B - 1) / B, B, 0, stream>>>(src, dst, dinv, norm, E);

    const long long spmmThreads = (long long)E * 32;
    const int spmmBlocks = (int)((spmmThreads + B - 1) / B);

    // ---- layer 1: agg = SpMM(X); h = relu(agg @ W1)
    zero_f32<<<4096, B, 0, stream>>>(agg, (long long)N * 128);
    spmm_scatter128<<<spmmBlocks, B, 0, stream>>>(X, src, dst, norm, agg, E);
    gemm_relu_wmma_lds<8><<<gemmBlocks, B, 0, stream>>>(agg, W1, h, N);

    // ---- layer 2: agg = SpMM(h); h = relu(agg @ W2)
    zero_f32<<<4096, B, 0, stream>>>(agg, (long long)N * 128);
    spmm_scatter128<<<spmmBlocks, B, 0, stream>>>(h, src, dst, norm, agg, E);
    gemm_relu_wmma_lds<8><<<gemmBlocks, B, 0, stream>>>(agg, W2, h, N);

    // ---- layer 3: agg = SpMM(h); out = relu(agg @ W3); softmax rows
    zero_f32<<<4096, B, 0, stream>>>(agg, (long long)N * 128);
    spmm_scatter128<<<spmmBlocks, B, 0, stream>>>(h, src, dst, norm, agg, E);
    gemm_relu_wmma_lds<4><<<gemmBlocks, B, 0, stream>>>(agg, W3, out, N);

    softmax64<<<(N + wavesPerBlock - 1) / wavesPerBlock, B, 0, stream>>>(out, N);
}